// NetworkNet_48430051229954
// MI455X (gfx1250) — compile-verified
//
#include <hip/hip_runtime.h>

typedef float v2f __attribute__((ext_vector_type(2)));
typedef float v8f __attribute__((ext_vector_type(8)));

#define IN_DIM 128
#define NET_DIM 32

// ---------------------------------------------------------------------------
// GEMM1: h = x @ W_gcn   (N x 128) @ (128 x 32) -> (N x 32), fp32 WMMA
// One wave32 computes a 16x32 output tile via V_WMMA_F32_16X16X4_F32.
// ---------------------------------------------------------------------------
__global__ void gcn_gemm1(const float* __restrict__ x, const float* __restrict__ W,
                          float* __restrict__ h, int n, int ntiles) {
  const int lane = threadIdx.x & 31;
  const int tile = blockIdx.x * (blockDim.x >> 5) + (threadIdx.x >> 5);
  if (tile >= ntiles) return;            // wave-uniform: EXEC stays all-ones
  const int m  = lane & 15;
  const int hi = lane >> 4;              // 0 or 1 (lane half)
  int row = tile * 16 + m;
  if (row >= n) row = n - 1;             // clamp loads; stores are guarded
  const float* xr = x + (long long)row * IN_DIM;

  v8f c0 = {}, c1 = {};
  for (int k0 = 0; k0 < IN_DIM; k0 += 4) {
    const int ka = k0 + 2 * hi;
    v2f a;  a.x = xr[ka];  a.y = xr[ka + 1];
    v2f b0, b1;
    b0.x = W[ka * NET_DIM + m];          b0.y = W[(ka + 1) * NET_DIM + m];
    b1.x = W[ka * NET_DIM + 16 + m];     b1.y = W[(ka + 1) * NET_DIM + 16 + m];
    c0 = __builtin_amdgcn_wmma_f32_16x16x4_f32(false, a, false, b0, (short)0, c0, false, false);
    c1 = __builtin_amdgcn_wmma_f32_16x16x4_f32(false, a, false, b1, (short)0, c1, false, false);
  }
#pragma unroll
  for (int v = 0; v < 8; ++v) {
    const int rr = tile * 16 + v + 8 * hi;
    if (rr < n) {
      h[(long long)rr * NET_DIM + m]      = c0[v];
      h[(long long)rr * NET_DIM + 16 + m] = c1[v];
    }
  }
}

// ---------------------------------------------------------------------------
// Degree / normalization
// ---------------------------------------------------------------------------
__global__ void init_deg(float* __restrict__ deg, int n) {
  int i = blockIdx.x * blockDim.x + threadIdx.x;
  if (i < n) deg[i] = 1.0f;              // self-loop contributes 1
}

__global__ void edge_deg(const int* __restrict__ col, float* __restrict__ deg, int e) {
  int i = blockIdx.x * blockDim.x + threadIdx.x;
  if (i < e) atomicAdd(&deg[col[i]], 1.0f);
}

__global__ void deg_to_dinv(float* __restrict__ deg, int n) {
  int i = blockIdx.x * blockDim.x + threadIdx.x;
  if (i < n) deg[i] = __frsqrt_rn(deg[i]);   // deg >= 1 always
}

// agg[i][c] = dinv[i]^2 * h[i][c]   (self-loop message, also zero-inits agg)
__global__ void self_loop_init(const float* __restrict__ h, const float* __restrict__ dinv,
                               float* __restrict__ agg, int n) {
  int t = blockIdx.x * blockDim.x + threadIdx.x;
  if (t < n * NET_DIM) {
    int i = t >> 5;                       // NET_DIM == 32
    float d = dinv[i];
    agg[t] = d * d * h[t];
  }
}

// ---------------------------------------------------------------------------
// Edge scatter: 32 lanes per edge (one lane per feature). Dominant kernel:
// one coalesced 128B h-row load + 32 global_atomic_add_f32 per edge.
// ---------------------------------------------------------------------------
__global__ void edge_scatter(const int* __restrict__ row, const int* __restrict__ col,
                             const float* __restrict__ h, const float* __restrict__ dinv,
                             float* __restrict__ agg, int e) {
  int t = blockIdx.x * blockDim.x + threadIdx.x;
  int edge = t >> 5;
  if (edge >= e) return;
  int c  = t & 31;
  int r  = row[edge];
  int cl = col[edge];
  float norm = dinv[r] * dinv[cl];
  atomicAdd(&agg[cl * NET_DIM + c], norm * h[r * NET_DIM + c]);
}

// ---------------------------------------------------------------------------
// GEMM2 fused: out = relu( relu(agg + b_gcn) @ W_dense + b_dense ), fp32 WMMA
// ---------------------------------------------------------------------------
__global__ void gcn_gemm2(const float* __restrict__ agg, const float* __restrict__ bg,
                          const float* __restrict__ Wd, const float* __restrict__ bd,
                          float* __restrict__ out, int n, int ntiles) {
  const int lane = threadIdx.x & 31;
  const int tile = blockIdx.x * (blockDim.x >> 5) + (threadIdx.x >> 5);
  if (tile >= ntiles) return;
  const int m  = lane & 15;
  const int hi = lane >> 4;
  int row = tile * 16 + m;
  if (row >= n) row = n - 1;
  const float* ar = agg + (long long)row * NET_DIM;

  v8f c0 = {}, c1 = {};
  for (int k0 = 0; k0 < NET_DIM; k0 += 4) {
    const int ka = k0 + 2 * hi;
    v2f a;
    a.x = fmaxf(ar[ka]     + bg[ka],     0.0f);
    a.y = fmaxf(ar[ka + 1] + bg[ka + 1], 0.0f);
    v2f b0, b1;
    b0.x = Wd[ka * NET_DIM + m];         b0.y = Wd[(ka + 1) * NET_DIM + m];
    b1.x = Wd[ka * NET_DIM + 16 + m];    b1.y = Wd[(ka + 1) * NET_DIM + 16 + m];
    c0 = __builtin_amdgcn_wmma_f32_16x16x4_f32(false, a, false, b0, (short)0, c0, false, false);
    c1 = __builtin_amdgcn_wmma_f32_16x16x4_f32(false, a, false, b1, (short)0, c1, false, false);
  }
#pragma unroll
  for (int v = 0; v < 8; ++v) {
    const int rr = tile * 16 + v + 8 * hi;
    if (rr < n) {
      out[(long long)rr * NET_DIM + m]      = fmaxf(c0[v] + bd[m],      0.0f);
      out[(long long)rr * NET_DIM + 16 + m] = fmaxf(c1[v] + bd[16 + m], 0.0f);
    }
  }
}

// ---------------------------------------------------------------------------
extern "C" void kernel_launch(void* const* d_in, const int* in_sizes, int n_in,
                              void* d_out, int out_size, void* d_ws, size_t ws_size,
                              hipStream_t stream) {
  const float* x      = (const float*)d_in[0];
  const int*   eidx   = (const int*)d_in[1];   // [2, E]: row = eidx[0..E), col = eidx[E..2E)
  const float* W_gcn  = (const float*)d_in[2];
  const float* b_gcn  = (const float*)d_in[3];
  const float* W_den  = (const float*)d_in[4];
  const float* b_den  = (const float*)d_in[5];
  float*       out    = (float*)d_out;

  const int N = in_sizes[0] / IN_DIM;
  const int E = in_sizes[1] / 2;

  float* ws   = (float*)d_ws;
  float* h    = ws;                         // N * 32
  float* agg  = ws + (size_t)N * NET_DIM;   // N * 32
  float* dinv = agg + (size_t)N * NET_DIM;  // N   (deg, then dinv in place)

  const int* erow = eidx;
  const int* ecol = eidx + E;

  const int ntiles = (N + 15) / 16;
  const int wavesPerBlock = 8;              // 256 threads = 8 wave32
  dim3 gemmGrid((ntiles + wavesPerBlock - 1) / wavesPerBlock);
  dim3 gemmBlock(32 * wavesPerBlock);

  // 1) h = x @ W_gcn (WMMA f32)
  gcn_gemm1<<<gemmGrid, gemmBlock, 0, stream>>>(x, W_gcn, h, N, ntiles);

  // 2) degree -> dinv
  init_deg<<<(N + 255) / 256, 256, 0, stream>>>(dinv, N);
  edge_deg<<<(E + 255) / 256, 256, 0, stream>>>(ecol, dinv, E);
  deg_to_dinv<<<(N + 255) / 256, 256, 0, stream>>>(dinv, N);

  // 3) agg = self-loop messages (also initializes agg)
  self_loop_init<<<((size_t)N * NET_DIM + 255) / 256, 256, 0, stream>>>(h, dinv, agg, N);

  // 4) edge scatter-add (dominant, atomic-bound)
  {
    long long threads = (long long)E * NET_DIM;
    edge_scatter<<<(unsigned)((threads + 255) / 256), 256, 0, stream>>>(erow, ecol, h, dinv, agg, E);
  }

  // 5) out = relu(relu(agg + b_gcn) @ W_dense + b_dense) (WMMA f32, fused)
  gcn_gemm2<<<gemmGrid, gemmBlock, 0, stream>>>(agg, b_gcn, W_den, b_den, out, N, ntiles);
}